// SSMCell_37752762532580
// MI455X (gfx1250) — compile-verified
//
#include <hip/hip_runtime.h>
#include <hip/hip_bf16.h>

#define D_INNER 8192
#define DT_RANK 8
#define D_STATE 16
#define D_MODEL 4096

typedef __attribute__((ext_vector_type(2))) float v2f;
typedef __attribute__((ext_vector_type(4))) float v4f;
typedef __attribute__((ext_vector_type(8))) float v8f;

// ---------------------------------------------------------------------------
// WMMA-based GEMV: out[M] = W[M x K] @ x[K]   (fp32, row-major W)
// One wave per 16 rows. Each lane streams a contiguous float4 of its row:
//   lanes 0-15  : W[row0+m][k .. k+3]
//   lanes 16-31 : W[row0+m][k+4 .. k+7]
// Two V_WMMA_F32_16X16X4_F32 per float4 (lo half, hi half). B is x[k'] loaded
// from LDS with the *same* slot addressing, broadcast across all 16 columns,
// so every column of D equals the GEMV partial sums (slot->k relabeling is
// consistent between A and B, which is all the dot product needs).
// W is read with non-temporal hints: 402 MB/call > 192 MB L2, zero reuse.
// Requires: K % 8 == 0, M % (16 * waves_per_block) == 0, blockDim.x = 256.
// ---------------------------------------------------------------------------
__global__ void gemv16_wmma_f32(const float* __restrict__ W,
                                const float* __restrict__ x,
                                float* __restrict__ out,
                                int K) {
    __shared__ __align__(16) float sx[8192];         // max K we use is 8192
    for (int k = threadIdx.x; k < K; k += blockDim.x) sx[k] = x[k];
    __syncthreads();

    const int lane = threadIdx.x & 31;
    const int wave = (blockIdx.x * blockDim.x + threadIdx.x) >> 5;
    const int row0 = wave << 4;                      // 16 rows per wave
    const int half = lane >> 4;                      // which float4 of the 8-K chunk
    const int m    = lane & 15;
    const int koff = half << 2;

    const float* wrow = W + (size_t)(row0 + m) * (size_t)K + koff;
    const float* bx   = sx + koff;

    v8f acc0 = {};
    v8f acc1 = {};
#pragma unroll 4
    for (int k = 0; k < K; k += 8) {
        v4f a = __builtin_nontemporal_load((const v4f*)(wrow + k));
        v4f b = *(const v4f*)(bx + k);
        v2f a0 = __builtin_shufflevector(a, a, 0, 1);
        v2f a1 = __builtin_shufflevector(a, a, 2, 3);
        v2f b0 = __builtin_shufflevector(b, b, 0, 1);
        v2f b1 = __builtin_shufflevector(b, b, 2, 3);
        acc0 = __builtin_amdgcn_wmma_f32_16x16x4_f32(
            false, a0, false, b0, (short)0, acc0, false, false);
        acc1 = __builtin_amdgcn_wmma_f32_16x16x4_f32(
            false, a1, false, b1, (short)0, acc1, false, false);
    }

    // C/D layout: VGPR v -> M=v on lanes 0-15, M=v+8 on lanes 16-31.
    // All N columns identical; lane 0 emits rows 0-7, lane 16 rows 8-15.
    if ((lane & 15) == 0) {
        float* o = out + row0 + (half << 3);
#pragma unroll
        for (int v = 0; v < 8; ++v) o[v] = acc0[v] + acc1[v];
    }
}

// ---------------------------------------------------------------------------
// Depthwise conv(4) over [b1, b2, xb, xb] + bias, then SiLU.
// ---------------------------------------------------------------------------
__global__ void conv_silu_kernel(const float* __restrict__ xz,
                                 const float* __restrict__ conv_buffer,
                                 const float* __restrict__ conv_w,
                                 const float* __restrict__ conv_b,
                                 float* __restrict__ x_conv) {
    int i = blockIdx.x * blockDim.x + threadIdx.x;
    if (i >= D_INNER) return;
    float xb = xz[i];                                // x_branch = xz[:D_INNER]
    float b1 = conv_buffer[i * 3 + 1];
    float b2 = conv_buffer[i * 3 + 2];
    float4 w = *(const float4*)(conv_w + (size_t)i * 4);
    float v  = b1 * w.x + b2 * w.y + xb * w.z + xb * w.w + conv_b[i];
    x_conv[i] = v / (1.0f + __expf(-v));             // silu
}

// ---------------------------------------------------------------------------
// Small GEMV: out[row] = W[row, :K] . x  for 40 rows (one block per row).
// ---------------------------------------------------------------------------
__global__ void gemv_small_kernel(const float* __restrict__ W,
                                  const float* __restrict__ x,
                                  float* __restrict__ out,
                                  int K) {
    __shared__ float red[256];
    const int row = blockIdx.x;
    const float* w = W + (size_t)row * (size_t)K;
    float s = 0.0f;
    for (int k = threadIdx.x; k < K; k += 256) s += w[k] * x[k];
    red[threadIdx.x] = s;
    __syncthreads();
    for (int off = 128; off > 0; off >>= 1) {
        if ((int)threadIdx.x < off) red[threadIdx.x] += red[threadIdx.x + off];
        __syncthreads();
    }
    if (threadIdx.x == 0) out[row] = red[0];
}

// ---------------------------------------------------------------------------
// Fused: dt = softplus(W_dt @ xp[:8] + b_dt); selective-scan state update;
// y = (h . C) + D*x_conv, gated by silu(z). Writes new_h directly to d_out.
// ---------------------------------------------------------------------------
__global__ void ssm_update_kernel(const float* __restrict__ xz,        // z at +D_INNER
                                  const float* __restrict__ x_conv,
                                  const float* __restrict__ xp,        // 40 floats
                                  const float* __restrict__ W_dt,
                                  const float* __restrict__ b_dt,
                                  const float* __restrict__ A_log,
                                  const float* __restrict__ ssm_state,
                                  const float* __restrict__ D_param,
                                  float* __restrict__ y,
                                  float* __restrict__ new_h) {
    __shared__ float sxp[DT_RANK + 2 * D_STATE];     // 40
    if (threadIdx.x < DT_RANK + 2 * D_STATE) sxp[threadIdx.x] = xp[threadIdx.x];
    __syncthreads();

    int i = blockIdx.x * blockDim.x + threadIdx.x;
    if (i >= D_INNER) return;

    // dt = softplus(W_dt[i,:] . xp[:8] + b_dt[i])
    v4f wd0 = *(const v4f*)(W_dt + (size_t)i * DT_RANK);
    v4f wd1 = *(const v4f*)(W_dt + (size_t)i * DT_RANK + 4);
    float acc = b_dt[i];
#pragma unroll
    for (int r = 0; r < 4; ++r) acc += wd0[r] * sxp[r];
#pragma unroll
    for (int r = 0; r < 4; ++r) acc += wd1[r] * sxp[4 + r];
    float dt = (acc > 20.0f) ? acc : log1pf(__expf(acc));

    float xc = x_conv[i];
    float ysum = 0.0f;
#pragma unroll
    for (int q = 0; q < 4; ++q) {
        v4f al = *(const v4f*)(A_log + (size_t)i * D_STATE + q * 4);
        v4f hs = *(const v4f*)(ssm_state + (size_t)i * D_STATE + q * 4);
        v4f hn;
#pragma unroll
        for (int s = 0; s < 4; ++s) {
            float A    = -__expf(al[s]);
            float Abar = __expf(dt * A);
            float B    = sxp[DT_RANK + q * 4 + s];
            float C    = sxp[DT_RANK + D_STATE + q * 4 + s];
            float h    = Abar * hs[s] + dt * B * xc;
            hn[s] = h;
            ysum += h * C;
        }
        *(v4f*)(new_h + (size_t)i * D_STATE + q * 4) = hn;
    }
    float yv = ysum + D_param[i] * xc;
    float z  = xz[D_INNER + i];
    yv *= z / (1.0f + __expf(-z));                   // * silu(z)
    y[i] = yv;
}

extern "C" void kernel_launch(void* const* d_in, const int* in_sizes, int n_in,
                              void* d_out, int out_size, void* d_ws, size_t ws_size,
                              hipStream_t stream) {
    const float* x           = (const float*)d_in[0];
    const float* ssm_state   = (const float*)d_in[1];
    const float* conv_buffer = (const float*)d_in[2];
    const float* W_in        = (const float*)d_in[3];
    const float* conv_w      = (const float*)d_in[4];
    const float* conv_b      = (const float*)d_in[5];
    const float* W_xp        = (const float*)d_in[6];
    const float* W_dt        = (const float*)d_in[7];
    const float* b_dt        = (const float*)d_in[8];
    const float* A_log       = (const float*)d_in[9];
    const float* D_param     = (const float*)d_in[10];
    const float* W_out       = (const float*)d_in[11];

    float* out   = (float*)d_out;                    // [0, 4096): out
    float* new_h = out + D_MODEL;                    // [4096, 4096+8192*16): new_h

    float* ws     = (float*)d_ws;
    float* xz     = ws;                              // 16384
    float* x_conv = ws + 2 * D_INNER;                // 8192
    float* xp     = ws + 3 * D_INNER;                // 40 (padded to 64)
    float* y      = ws + 3 * D_INNER + 64;           // 8192

    // 1) xz = W_in @ x              (268 MB stream — dominant)
    //    256 threads = 8 waves -> 128 rows/block; 16384 rows -> 128 blocks
    gemv16_wmma_f32<<<(2 * D_INNER) / 128, 256, 0, stream>>>(W_in, x, xz, D_MODEL);

    // 2) causal conv + SiLU
    conv_silu_kernel<<<D_INNER / 256, 256, 0, stream>>>(xz, conv_buffer, conv_w,
                                                        conv_b, x_conv);

    // 3) xp = W_xp @ x_conv         (40 x 8192)
    gemv_small_kernel<<<DT_RANK + 2 * D_STATE, 256, 0, stream>>>(W_xp, x_conv, xp,
                                                                 D_INNER);

    // 4) dt / state update / gated y ; new_h -> d_out
    ssm_update_kernel<<<D_INNER / 256, 256, 0, stream>>>(xz, x_conv, xp, W_dt, b_dt,
                                                         A_log, ssm_state, D_param,
                                                         y, new_h);

    // 5) out = W_out @ y            (134 MB stream)
    gemv16_wmma_f32<<<D_MODEL / 128, 256, 0, stream>>>(W_out, y, out, D_INNER);
}